// PagedAttention_31379031064676
// MI455X (gfx1250) — compile-verified
//
#include <hip/hip_runtime.h>

// ---------------------------------------------------------------------------
// Paged-attention decode for MI455X (gfx1250), wave32, WMMA f32 16x16x4.
// B=16, H=16, S=4096, D=128, P=32768.  HBM-bound: ~1.07 GB of K/V gather
// traffic -> ~46us floor @ 23.3 TB/s.  Full-f32 math (matches reference).
//
// Round 2: batch tile loads into register arrays so the compiler issues all
// global_load_b64 back-to-back (MLP=32) instead of the round-1 serialized
// load -> s_wait_loadcnt 0 -> wmma chain.
// ---------------------------------------------------------------------------

typedef __attribute__((ext_vector_type(2))) float v2f;
typedef __attribute__((ext_vector_type(8))) float v8f;

#if defined(__HIP_DEVICE_COMPILE__)
#if !__has_builtin(__builtin_amdgcn_wmma_f32_16x16x4_f32)
#error "toolchain missing __builtin_amdgcn_wmma_f32_16x16x4_f32"
#endif
#endif

#define BATCH    16
#define HEADS    16
#define SEQ      4096
#define HDIM     128
#define NPAGES   32768

#define WAVES_PER_BLOCK 8
#define TILE            16
#define TOK_PER_WAVE    (SEQ / WAVES_PER_BLOCK)      // 512
#define TILES_PER_WAVE  (TOK_PER_WAVE / TILE)        // 32
#define NCHUNK          (HDIM / 4)                   // 32 WMMA k-chunks

__global__ __launch_bounds__(WAVES_PER_BLOCK * 32)
void paged_attn_f32_wmma(const float* __restrict__ q,
                         const float* __restrict__ kv,
                         const int*   __restrict__ k_idx,
                         const int*   __restrict__ v_idx,
                         float*       __restrict__ out)
{
    __shared__ __align__(16) float sh_q[HDIM];
    __shared__ __align__(16) float sh_part[WAVES_PER_BLOCK][HDIM];
    __shared__ float sh_m[WAVES_PER_BLOCK];
    __shared__ float sh_l[WAVES_PER_BLOCK];

    const int bh   = blockIdx.x;          // b*HEADS + h
    const int b    = bh / HEADS;
    const int h    = bh % HEADS;
    const int tid  = threadIdx.x;
    const int wave = tid >> 5;
    const int lane = tid & 31;
    const int n    = lane & 15;           // matrix column / token within tile
    const int koff = (lane >> 4) << 1;    // 0 for lanes 0-15, 2 for lanes 16-31

    // Stage q row (512 B) into LDS; broadcast-read later as WMMA A operand.
    if (tid < HDIM) sh_q[tid] = q[(size_t)bh * HDIM + tid];
    __syncthreads();

    const float* kc = kv;                                       // [P,H,D]
    const float* vc = kv + (size_t)NPAGES * HEADS * HDIM;       // [P,H,D]
    const int*   ki = k_idx + (size_t)b * SEQ;
    const int*   vi = v_idx + (size_t)b * SEQ;
    const float scale = 0.08838834764831845f;                   // 1/sqrt(128)

    float m = -1e30f;
    float l = 0.0f;
    v8f acc[8];
#pragma unroll
    for (int g = 0; g < 8; ++g) acc[g] = (v8f){0.f,0.f,0.f,0.f,0.f,0.f,0.f,0.f};

    const int s0 = wave * TOK_PER_WAVE;

    for (int t = 0; t < TILES_PER_WAVE; ++t) {
        const int sb = s0 + t * TILE;

        // Lane r (<16) owns token r's page ids; broadcast via shfl.
        int pk = (lane < 16) ? ki[sb + lane] : 0;
        int pv = (lane < 16) ? vi[sb + lane] : 0;

        // Prefetch next tile's rows (global_prefetch_b8) to hide gather latency.
        if (t + 1 < TILES_PER_WAVE) {
            int nt = sb + TILE + n;
            if (lane < 16) {
                int np = ki[nt];
                __builtin_prefetch(kc + ((size_t)np * HEADS + h) * HDIM, 0, 3);
            } else {
                int np = vi[nt];
                __builtin_prefetch(vc + ((size_t)np * HEADS + h) * HDIM, 0, 3);
            }
        }

        // Per-lane K-row pointer: lane n and n+16 both point at token n's row.
        const int my_pk = __shfl(pk, n, 32);
        const float* krow = kc + ((size_t)my_pk * HEADS + h) * HDIM;

        // ---- Phase 1: batch-load all 32 K chunk operands (independent regs,
        //      so the 32 global_load_b64 issue back-to-back). ----
        v2f kreg[NCHUNK];
#pragma unroll
        for (int c = 0; c < NCHUNK; ++c)
            kreg[c] = *(const v2f*)(krow + c * 4 + koff);

        // ---- Phase 2: scores = q . K^T : 32 x V_WMMA_F32_16X16X4_F32 ----
        v8f sc = (v8f){0.f,0.f,0.f,0.f,0.f,0.f,0.f,0.f};
#pragma unroll
        for (int c = 0; c < NCHUNK; ++c) {
            v2f a = *(const v2f*)(sh_q + c * 4 + koff);  // rows replicated
            sc = __builtin_amdgcn_wmma_f32_16x16x4_f32(
                     false, a, false, kreg[c], (short)0, sc, false, false);
        }
        // All D rows identical -> every lane holds score[lane&15] in sc[0].
        float s = sc[0] * scale;

        // ---- online softmax over the 16-token tile ----
        float tmax = s;
#pragma unroll
        for (int off = 8; off >= 1; off >>= 1)
            tmax = fmaxf(tmax, __shfl_xor(tmax, off, 32));
        float m_new = fmaxf(m, tmax);
        float alpha = __expf(m - m_new);
        float p     = __expf(s - m_new);
        float psum  = p;
#pragma unroll
        for (int off = 8; off >= 1; off >>= 1)
            psum += __shfl_xor(psum, off, 32);
        l = l * alpha + psum;
        m = m_new;
#pragma unroll
        for (int g = 0; g < 8; ++g) acc[g] *= alpha;

        // ---- acc += P . V : 4 k-chunks x 8 dim-groups, loads batched per kb ----
#pragma unroll
        for (int kb = 0; kb < TILE; kb += 4) {
            // A: probs chunk, replicated rows (lanes<16: p[kb],p[kb+1]; else p[kb+2],p[kb+3])
            float a0 = __shfl(p, kb + koff, 32);
            float a1 = __shfl(p, kb + koff + 1, 32);
            v2f a = {a0, a1};
            // Row pointers for the two V rows this half-wave consumes.
            int r0 = __shfl(pv, kb + koff, 32);
            int r1 = __shfl(pv, kb + koff + 1, 32);
            const float* v0p = vc + ((size_t)r0 * HEADS + h) * HDIM;
            const float* v1p = vc + ((size_t)r1 * HEADS + h) * HDIM;

            // Phase 1: batch the 16 b32 gathers for this kb chunk.
            v2f vreg[8];
#pragma unroll
            for (int g = 0; g < 8; ++g) {
                vreg[g].x = v0p[g * 16 + n];
                vreg[g].y = v1p[g * 16 + n];
            }
            // Phase 2: 8 WMMAs accumulating into the output dims.
#pragma unroll
            for (int g = 0; g < 8; ++g) {
                acc[g] = __builtin_amdgcn_wmma_f32_16x16x4_f32(
                             false, a, false, vreg[g], (short)0, acc[g], false, false);
            }
        }
    }

    // ---- per-wave partial (unnormalized) out + (m, l) to LDS ----
    if (lane < 16) {
#pragma unroll
        for (int g = 0; g < 8; ++g)
            sh_part[wave][g * 16 + lane] = acc[g][0];   // D[0, n] = dim g*16+n
    }
    if (lane == 0) { sh_m[wave] = m; sh_l[wave] = l; }
    __syncthreads();

    // ---- cross-wave combine: thread tid -> output dim (tid < 128) ----
    if (tid < HDIM) {
        float mg = sh_m[0];
#pragma unroll
        for (int w = 1; w < WAVES_PER_BLOCK; ++w) mg = fmaxf(mg, sh_m[w]);
        float L = 0.0f, O = 0.0f;
#pragma unroll
        for (int w = 0; w < WAVES_PER_BLOCK; ++w) {
            float aw = __expf(sh_m[w] - mg);
            L += aw * sh_l[w];
            O += aw * sh_part[w][tid];
        }
        out[(size_t)bh * HDIM + tid] = O / L;
    }
}

extern "C" void kernel_launch(void* const* d_in, const int* in_sizes, int n_in,
                              void* d_out, int out_size, void* d_ws, size_t ws_size,
                              hipStream_t stream) {
    (void)in_sizes; (void)n_in; (void)out_size; (void)d_ws; (void)ws_size;
    const float* q  = (const float*)d_in[0];
    const float* kv = (const float*)d_in[1];
    const int*   ki = (const int*)d_in[2];
    const int*   vi = (const int*)d_in[3];
    float* out = (float*)d_out;

    dim3 grid(BATCH * HEADS);              // one workgroup per (b,h)
    dim3 block(WAVES_PER_BLOCK * 32);      // 8 wave32s
    hipLaunchKernelGGL(paged_attn_f32_wmma, grid, block, 0, stream,
                       q, kv, ki, vi, out);
}